// GNN_68487548502670
// MI455X (gfx1250) — compile-verified
//
#include <hip/hip_runtime.h>
#include <math.h>
#include <stdint.h>

#define NEG_INF  -9000000000000000.0f
#define ALPHA    0.1f
#define N_ROWS   4096
#define D_DIM    2048

// Block tile 64(M) x 128(N), K-chunk 32. 256 threads = 8 waves, wave grid 2(M) x 4(N),
// each wave computes a 32x32 macro-tile = four 16x16 WMMA tiles (2 A-frags x 2 B-frags).
#define BM 64
#define BN 128
#define BK 32
#define AS_S   36   // As row stride (floats): BK+4, 16B-aligned rows (36*4=144=9*16)
#define BS_NT 136   // Bs [k][n] stride: BN+8 -> lo/hi lane halves hit disjoint banks
#define BS_BT  36   // Bs [n][k] stride

#define AS_FLOATS    (BM * AS_S)    // 2304
#define BS_NT_FLOATS (BK * BS_NT)   // 4352
#define BS_BT_FLOATS (BN * BS_BT)   // 4608

typedef float v8f __attribute__((ext_vector_type(8)));
typedef float v2f __attribute__((ext_vector_type(2)));

__device__ __forceinline__ v8f wmma_f32(v2f a, v2f b, v8f c) {
    // V_WMMA_F32_16X16X4_F32 : D = A(16x4) * B(4x16) + C(16x16)
    return __builtin_amdgcn_wmma_f32_16x16x4_f32(
        false, a, false, b, (short)0, c, false, false);
}

// CDNA5 async global->LDS copy (ASYNCcnt). GVS mode: saddr = 64-bit SGPR base,
// vaddr = 32-bit byte offset, dst VGPR = LDS byte offset within our allocation.
__device__ __forceinline__ void async_b128(uint32_t lds_off, const void* base, uint32_t goff) {
    asm volatile("global_load_async_to_lds_b128 %0, %1, %2"
                 :: "v"(lds_off), "v"(goff), "s"(base)
                 : "memory");
}
__device__ __forceinline__ void wait_async0() {
    asm volatile("s_wait_asynccnt 0x0" ::: "memory");
}

// Stage one K-chunk: As[row][k] (64x32), Bs either [n][k] (B row-major [N][K], A*B^T)
// or [k][n] (B row-major [K][N]). All pieces are 16B, 16B-aligned in LDS and global.
template <bool B_TRANS>
__device__ __forceinline__ void issue_chunk(const float* __restrict__ A, int lda,
                                            const float* __restrict__ B, int ldb,
                                            int mBlock, int nBlock, int k0,
                                            uint32_t asBase, uint32_t bsBase) {
    const int tid = threadIdx.x;
    #pragma unroll
    for (int i = 0; i < 2; ++i) {               // 64 rows x 8 float4 = 512 pieces
        int lin = tid + i * 256;
        int row = lin >> 3;
        int kc  = (lin & 7) << 2;
        async_b128(asBase + (uint32_t)(row * AS_S + kc) * 4u,
                   A, (uint32_t)((mBlock + row) * lda + k0 + kc) * 4u);
    }
    if (B_TRANS) {
        #pragma unroll
        for (int i = 0; i < 4; ++i) {           // 128 rows x 8 float4 = 1024 pieces
            int lin = tid + i * 256;
            int j   = lin >> 3;
            int kc  = (lin & 7) << 2;
            async_b128(bsBase + (uint32_t)(j * BS_BT + kc) * 4u,
                       B, (uint32_t)((nBlock + j) * ldb + k0 + kc) * 4u);
        }
    } else {
        #pragma unroll
        for (int i = 0; i < 4; ++i) {           // 32 k-rows x 32 float4 = 1024 pieces
            int lin = tid + i * 256;
            int kr  = lin >> 5;
            int nc  = (lin & 31) << 2;
            async_b128(bsBase + (uint32_t)(kr * BS_NT + nc) * 4u,
                       B, (uint32_t)((k0 + kr) * ldb + nBlock + nc) * 4u);
        }
    }
}

template <bool B_TRANS>
__device__ __forceinline__ void compute_chunk(const float* __restrict__ Asp,
                                              const float* __restrict__ Bsp,
                                              v8f& acc00, v8f& acc01,
                                              v8f& acc10, v8f& acc11) {
    const int lane = threadIdx.x & 31;
    const int wave = threadIdx.x >> 5;
    const int mw = (wave & 1) * 32;
    const int nw = (wave >> 1) * 32;
    const int hi = lane >> 4;
    const int lo = lane & 15;
    #pragma unroll
    for (int kk = 0; kk < BK; kk += 4) {
        const int ka = kk + 2 * hi;   // even -> 8B-aligned v2f
        v2f a0 = *(const v2f*)(&Asp[(mw + lo) * AS_S + ka]);
        v2f a1 = *(const v2f*)(&Asp[(mw + 16 + lo) * AS_S + ka]);
        v2f b0, b1;
        if (B_TRANS) {
            b0 = *(const v2f*)(&Bsp[(nw + lo) * BS_BT + ka]);
            b1 = *(const v2f*)(&Bsp[(nw + 16 + lo) * BS_BT + ka]);
        } else {
            b0.x = Bsp[ka * BS_NT + nw + lo];
            b0.y = Bsp[(ka + 1) * BS_NT + nw + lo];
            b1.x = Bsp[ka * BS_NT + nw + 16 + lo];
            b1.y = Bsp[(ka + 1) * BS_NT + nw + 16 + lo];
        }
        acc00 = wmma_f32(a0, b0, acc00);
        acc01 = wmma_f32(a0, b1, acc01);
        acc10 = wmma_f32(a1, b0, acc10);
        acc11 = wmma_f32(a1, b1, acc11);
    }
}

// Double-buffered async pipeline: wait(chunk c) -> barrier -> prefetch(chunk c+1)
// into the other buffer -> WMMA on chunk c. Buffer p is overwritten only at
// chunk c+2's issue, which all waves reach only after the barrier following
// their chunk-c reads.
template <bool B_TRANS>
__device__ __forceinline__ void gemm_main(const float* __restrict__ A, int lda,
                                          const float* __restrict__ B, int ldb,
                                          int K, int mBlock, int nBlock,
                                          float* As, float* Bs, int bsFloats,
                                          v8f& acc00, v8f& acc01,
                                          v8f& acc10, v8f& acc11) {
    const uint32_t asAddr  = (uint32_t)(uintptr_t)As;
    const uint32_t bsAddr  = (uint32_t)(uintptr_t)Bs;
    const uint32_t asBytes = AS_FLOATS * 4u;
    const uint32_t bsBytes = (uint32_t)bsFloats * 4u;
    const int NC = K / BK;
    issue_chunk<B_TRANS>(A, lda, B, ldb, mBlock, nBlock, 0, asAddr, bsAddr);
    for (int c = 0; c < NC; ++c) {
        const int p0 = c & 1;
        wait_async0();
        __syncthreads();
        if (c + 1 < NC) {
            const int p1 = p0 ^ 1;
            issue_chunk<B_TRANS>(A, lda, B, ldb, mBlock, nBlock, (c + 1) * BK,
                                 asAddr + p1 * asBytes, bsAddr + p1 * bsBytes);
        }
        compute_chunk<B_TRANS>(As + p0 * AS_FLOATS, Bs + p0 * bsFloats,
                               acc00, acc01, acc10, acc11);
    }
}

// ---------------- kernel 1: row L2 norms of x ----------------
__global__ __launch_bounds__(256) void rownorm_kernel(const float* __restrict__ x,
                                                      float* __restrict__ lens) {
    __shared__ float red[256];
    const int row = blockIdx.x;
    float s = 0.0f;
    for (int c = threadIdx.x; c < D_DIM; c += 256) {
        float v = x[(size_t)row * D_DIM + c];
        s += v * v;
    }
    red[threadIdx.x] = s;
    __syncthreads();
    for (int off = 128; off > 0; off >>= 1) {
        if (threadIdx.x < off) red[threadIdx.x] += red[threadIdx.x + off];
        __syncthreads();
    }
    if (threadIdx.x == 0) lens[row] = sqrtf(red[0]);
}

// ---------------- kernel 2: h = x @ W ----------------
__global__ __launch_bounds__(256) void gemm_xw_kernel(const float* __restrict__ x,
                                                      const float* __restrict__ W,
                                                      float* __restrict__ h) {
    __shared__ __align__(16) float As[2 * AS_FLOATS];
    __shared__ __align__(16) float Bs[2 * BS_NT_FLOATS];
    v8f acc00 = {}, acc01 = {}, acc10 = {}, acc11 = {};
    const int mBlock = blockIdx.y * BM;
    const int nBlock = blockIdx.x * BN;
    gemm_main<false>(x, D_DIM, W, D_DIM, D_DIM, mBlock, nBlock,
                     As, Bs, BS_NT_FLOATS, acc00, acc01, acc10, acc11);

    const int lane = threadIdx.x & 31;
    const int wave = threadIdx.x >> 5;
    const int mA = mBlock + (wave & 1) * 32 + 8 * (lane >> 4);
    const int nA = nBlock + (wave >> 1) * 32 + (lane & 15);
    #pragma unroll
    for (int r = 0; r < 8; ++r) {
        h[(size_t)(mA + r) * D_DIM + nA]           = acc00[r];
        h[(size_t)(mA + r) * D_DIM + nA + 16]      = acc01[r];
        h[(size_t)(mA + 16 + r) * D_DIM + nA]      = acc10[r];
        h[(size_t)(mA + 16 + r) * D_DIM + nA + 16] = acc11[r];
    }
}

// ---------------- kernel 3: s1 = h@a1, s2 = h@a2 ----------------
__global__ __launch_bounds__(256) void s12_kernel(const float* __restrict__ h,
                                                  const float* __restrict__ a,
                                                  float* __restrict__ s1,
                                                  float* __restrict__ s2) {
    __shared__ float r1[256];
    __shared__ float r2[256];
    const int row = blockIdx.x;
    float acc1 = 0.0f, acc2 = 0.0f;
    for (int c = threadIdx.x; c < D_DIM; c += 256) {
        float hv = h[(size_t)row * D_DIM + c];
        acc1 += hv * a[c];
        acc2 += hv * a[D_DIM + c];
    }
    r1[threadIdx.x] = acc1;
    r2[threadIdx.x] = acc2;
    __syncthreads();
    for (int off = 128; off > 0; off >>= 1) {
        if (threadIdx.x < off) {
            r1[threadIdx.x] += r1[threadIdx.x + off];
            r2[threadIdx.x] += r2[threadIdx.x + off];
        }
        __syncthreads();
    }
    if (threadIdx.x == 0) {
        s1[row] = r1[0];
        s2[row] = r2[0];
    }
}

// ---------------- kernel 4: x@x.T fused into masked logits P ----------------
__global__ __launch_bounds__(256) void xxt_logits_kernel(const float* __restrict__ x,
                                                         const float* __restrict__ adj,
                                                         const float* __restrict__ lens,
                                                         const float* __restrict__ s1,
                                                         const float* __restrict__ s2,
                                                         float* __restrict__ P) {
    __shared__ __align__(16) float As[2 * AS_FLOATS];
    __shared__ __align__(16) float Bs[2 * BS_BT_FLOATS];
    v8f acc00 = {}, acc01 = {}, acc10 = {}, acc11 = {};
    const int mBlock = blockIdx.y * BM;
    const int nBlock = blockIdx.x * BN;
    gemm_main<true>(x, D_DIM, x, D_DIM, D_DIM, mBlock, nBlock,
                    As, Bs, BS_BT_FLOATS, acc00, acc01, acc10, acc11);

    const int lane = threadIdx.x & 31;
    const int wave = threadIdx.x >> 5;
    const int mA = mBlock + (wave & 1) * 32 + 8 * (lane >> 4);
    const int nA = nBlock + (wave >> 1) * 32 + (lane & 15);
    const int nB = nA + 16;
    const float lnA = lens[nA], lnB = lens[nB];
    const float s2A = s2[nA],  s2B = s2[nB];
    #pragma unroll
    for (int r = 0; r < 8; ++r) {
        #pragma unroll
        for (int half = 0; half < 2; ++half) {
            const int m = mA + half * 16 + r;
            const float lm  = lens[m];
            const float s1m = s1[m];
            const float vA = half ? acc10[r] : acc00[r];
            const float vB = half ? acc11[r] : acc01[r];
            {
                float cosv = vA / (lm * lnA);
                float na   = adj[(size_t)m * N_ROWS + nA] * cosv;
                float t    = s1m + s2A;
                float e    = (t > 0.0f) ? t : ALPHA * t;
                P[(size_t)m * N_ROWS + nA] = (na > 0.0f) ? e : NEG_INF;
            }
            {
                float cosv = vB / (lm * lnB);
                float na   = adj[(size_t)m * N_ROWS + nB] * cosv;
                float t    = s1m + s2B;
                float e    = (t > 0.0f) ? t : ALPHA * t;
                P[(size_t)m * N_ROWS + nB] = (na > 0.0f) ? e : NEG_INF;
            }
        }
    }
}

// ---------------- kernel 5: row softmax of P (in place) ----------------
__global__ __launch_bounds__(256) void softmax_kernel(float* __restrict__ P) {
    __shared__ float red[256];
    const int row = blockIdx.x;
    float* __restrict__ p = P + (size_t)row * N_ROWS;

    float m = -3.4e38f;
    for (int c = threadIdx.x; c < N_ROWS; c += 256) m = fmaxf(m, p[c]);
    red[threadIdx.x] = m;
    __syncthreads();
    for (int off = 128; off > 0; off >>= 1) {
        if (threadIdx.x < off) red[threadIdx.x] = fmaxf(red[threadIdx.x], red[threadIdx.x + off]);
        __syncthreads();
    }
    const float rowmax = red[0];
    __syncthreads();

    float s = 0.0f;
    for (int c = threadIdx.x; c < N_ROWS; c += 256) {
        float e = __expf(p[c] - rowmax);
        p[c] = e;
        s += e;
    }
    red[threadIdx.x] = s;
    __syncthreads();
    for (int off = 128; off > 0; off >>= 1) {
        if (threadIdx.x < off) red[threadIdx.x] += red[threadIdx.x + off];
        __syncthreads();
    }
    const float inv = 1.0f / red[0];
    __syncthreads();

    for (int c = threadIdx.x; c < N_ROWS; c += 256) p[c] *= inv;
}

// ---------------- kernel 6: out = elu(P @ h) ----------------
__global__ __launch_bounds__(256) void ph_elu_kernel(const float* __restrict__ P,
                                                     const float* __restrict__ h,
                                                     float* __restrict__ out) {
    __shared__ __align__(16) float As[2 * AS_FLOATS];
    __shared__ __align__(16) float Bs[2 * BS_NT_FLOATS];
    v8f acc00 = {}, acc01 = {}, acc10 = {}, acc11 = {};
    const int mBlock = blockIdx.y * BM;
    const int nBlock = blockIdx.x * BN;
    gemm_main<false>(P, N_ROWS, h, D_DIM, N_ROWS, mBlock, nBlock,
                     As, Bs, BS_NT_FLOATS, acc00, acc01, acc10, acc11);

    const int lane = threadIdx.x & 31;
    const int wave = threadIdx.x >> 5;
    const int mA = mBlock + (wave & 1) * 32 + 8 * (lane >> 4);
    const int nA = nBlock + (wave >> 1) * 32 + (lane & 15);
    #pragma unroll
    for (int r = 0; r < 8; ++r) {
        float v00 = acc00[r], v01 = acc01[r], v10 = acc10[r], v11 = acc11[r];
        v00 = (v00 > 0.0f) ? v00 : expm1f(v00);
        v01 = (v01 > 0.0f) ? v01 : expm1f(v01);
        v10 = (v10 > 0.0f) ? v10 : expm1f(v10);
        v11 = (v11 > 0.0f) ? v11 : expm1f(v11);
        out[(size_t)(mA + r) * D_DIM + nA]           = v00;
        out[(size_t)(mA + r) * D_DIM + nA + 16]      = v01;
        out[(size_t)(mA + 16 + r) * D_DIM + nA]      = v10;
        out[(size_t)(mA + 16 + r) * D_DIM + nA + 16] = v11;
    }
}

extern "C" void kernel_launch(void* const* d_in, const int* in_sizes, int n_in,
                              void* d_out, int out_size, void* d_ws, size_t ws_size,
                              hipStream_t stream) {
    const float* x   = (const float*)d_in[0];  // (4096, 2048)
    const float* adj = (const float*)d_in[1];  // (4096, 4096)
    const float* W   = (const float*)d_in[2];  // (2048, 2048)
    const float* a   = (const float*)d_in[3];  // (4096,) flat
    float* out = (float*)d_out;                // (4096, 2048)

    // Workspace layout: h (32MB) | P (64MB) | lens | s1 | s2
    char* ws = (char*)d_ws;
    float* h    = (float*)(ws);
    float* P    = (float*)(ws + (size_t)N_ROWS * D_DIM * sizeof(float));
    float* lens = (float*)(ws + (size_t)N_ROWS * D_DIM * sizeof(float)
                              + (size_t)N_ROWS * N_ROWS * sizeof(float));
    float* s1   = lens + N_ROWS;
    float* s2   = s1 + N_ROWS;

    rownorm_kernel<<<N_ROWS, 256, 0, stream>>>(x, lens);

    dim3 gXW(D_DIM / BN, N_ROWS / BM);   // (16, 64)
    gemm_xw_kernel<<<gXW, 256, 0, stream>>>(x, W, h);

    s12_kernel<<<N_ROWS, 256, 0, stream>>>(h, a, s1, s2);

    dim3 gXXT(N_ROWS / BN, N_ROWS / BM); // (32, 64)
    xxt_logits_kernel<<<gXXT, 256, 0, stream>>>(x, adj, lens, s1, s2, P);

    softmax_kernel<<<N_ROWS, 256, 0, stream>>>(P);

    dim3 gPH(D_DIM / BN, N_ROWS / BM);   // (16, 64)
    ph_elu_kernel<<<gPH, 256, 0, stream>>>(P, h, out);
}